// KalmanFilter_28166395527563
// MI455X (gfx1250) — compile-verified
//
#include <hip/hip_runtime.h>
#include <hip/hip_bf16.h>

// ---------------------------------------------------------------------------
// Kalman rollout on MI455X (gfx1250, wave32) via blocked parallel scan.
//
//   s_t = A s_{t-1} + C u_t          (t = 0..T-1, s_{-1} = R(data0-mean))
//   out = S @ B^T + mean
//
// Decomposition: t = b*KB + j, KB=64 -> NB=512 blocks, superblocks K2=32.
//   y_{b,j}  : zero-init within-block scans (stride-2 sequential GEMM chain)
//   w_b      = y_{b,KB-1}
//   v_{B,i}  = sum_q M^q w_{B,i-q}   (M = A^KB, WMMA triangular kernel)
//   e_{B+1}  = M^K2 e_B + v_{B,K2-1} (tiny LDS scan)
//   c_b      = M^i e_B + v_{B,i-1}   (block carries)
//   Y[t]    += A^{j+1} c_b           (WMMA fixup), out = Y @ B^T + mean
//
// fp32 WMMA (v_wmma_f32_16x16x4_f32) throughout; each wave computes a 16x64
// strip (4 accumulators, A fragment loaded once per 4 WMMAs). The output GEMM
// stages its B-strip into LDS with global_load_async_to_lds_b128 (ASYNCcnt)
// so the 8 waves of a block share one copy instead of 8 L2 reads.
// ---------------------------------------------------------------------------

#define TSTEPS 32768
#define DZ 512
#define DX 512
#define DU 256
#define KB 64            // timesteps per block
#define NB (TSTEPS / KB) // 512 blocks
#define K2 32            // blocks per superblock
#define NB2 (NB / K2)    // 16 superblocks
#define ROWP 132         // padded LDS row (mod-64 = 4 -> conflict-free b64 reads)

typedef float v2f __attribute__((ext_vector_type(2)));
typedef float v8f __attribute__((ext_vector_type(8)));

// ---- WMMA strip helpers (16 x (NT*16) output per wave) --------------------
// Fragment layouts per CDNA5 ISA 7.12.2:
//   A 16x4 f32 : lanes 0-15 rows M=0..15 (K=0,1), lanes 16-31 same rows (K=2,3)
//   B 4x16 f32 : lane&15 = column N, half selects K-pair (mirrors A)
//   C/D        : VGPR v -> row M = v + 8*half, column N = lane&15

template <int NT>
__device__ __forceinline__ void wmma_bt(v8f* acc,
    const float* __restrict__ aTile, long lda,
    const float* __restrict__ bTile, long ldb,
    int kdim, int r, int koff)
{
  const float* ar = aTile + (long)r * lda + koff;
  const float* br = bTile + (long)r * ldb + koff;
#pragma unroll 2
  for (int k = 0; k < kdim; k += 4) {
    v2f a = *(const v2f*)(ar + k);
#pragma unroll
    for (int t = 0; t < NT; ++t) {
      v2f b = *(const v2f*)(br + (long)t * 16 * ldb + k);
      acc[t] = __builtin_amdgcn_wmma_f32_16x16x4_f32(false, a, false, b,
                                                     (short)0, acc[t],
                                                     false, false);
    }
  }
}

// acc[t] += Atile(16 x kdim) * Q(kdim x ncols, row stride ldq) cols [n0+16t)
template <int NT>
__device__ __forceinline__ void wmma_bn(v8f* acc,
    const float* __restrict__ aTile, long lda,
    const float* __restrict__ qTile, long ldq,
    int kdim, int r, int koff)
{
  const float* ar = aTile + (long)r * lda + koff;
  const float* qc = qTile + r;
#pragma unroll 2
  for (int k = 0; k < kdim; k += 4) {
    v2f a = *(const v2f*)(ar + k);
#pragma unroll
    for (int t = 0; t < NT; ++t) {
      v2f b;
      b.x = qc[(long)(k + koff) * ldq + 16 * t];
      b.y = qc[(long)(k + koff + 1) * ldq + 16 * t];
      acc[t] = __builtin_amdgcn_wmma_f32_16x16x4_f32(false, a, false, b,
                                                     (short)0, acc[t],
                                                     false, false);
    }
  }
}

template <int NT>
__device__ __forceinline__ void store_strip(float* __restrict__ outTile,
                                            long ldo, const v8f* acc,
                                            int r, int half)
{
#pragma unroll
  for (int t = 0; t < NT; ++t)
#pragma unroll
    for (int v = 0; v < 8; ++v)
      outTile[(long)(v + 8 * half) * ldo + 16 * t + r] = acc[t][v];
}

__device__ __forceinline__ void zero4(v8f* acc)
{
#pragma unroll
  for (int t = 0; t < 4; ++t)
    acc[t] = (v8f){0.f, 0.f, 0.f, 0.f, 0.f, 0.f, 0.f, 0.f};
}

// ---- batched square-matrix multiply: Out_i = P_i * Q ----------------------
// Power tables (A^(2^k) doubling) and AC = A*C. Q in (k,n) layout.
// jobs = (DZ/16) * (ncols/64); grid.x = jobs/8, grid.z = batch.
__global__ void mm_batch_kernel(const float* __restrict__ P, long pStride,
                                const float* __restrict__ Q,
                                float* __restrict__ Out, long oStride,
                                int kdim, int ncols)
{
  int wave = threadIdx.x >> 5, lane = threadIdx.x & 31;
  int r = lane & 15, half = lane >> 4, koff = 2 * half;
  int ngrp = ncols >> 6;
  int job = blockIdx.x * 8 + wave;
  int tm = job / ngrp, n0 = (job % ngrp) * 64;
  const float* Pi = P + (long)blockIdx.z * pStride;
  float* Oi = Out + (long)blockIdx.z * oStride;

  v8f acc[4]; zero4(acc);
  wmma_bn<4>(acc, Pi + (long)tm * 16 * kdim, kdim, Q + n0, ncols,
             kdim, r, koff);
  store_strip<4>(Oi + (long)tm * 16 * ncols + n0, ncols, acc, r, half);
}

// ---- stride-2 scan chain step ---------------------------------------------
// Yout(b,:) = [kPrev] Yprev(b,:) @ Wp^T + U1(b,:) @ W1^T + U2(b,:) @ W2^T
// Rows indexed by block b (stride KB*DZ / KB*DU between consecutive b).
// jobs = 32 m-tiles * 8 n-groups = 256; grid.x = 32.
__global__ void chain_step_kernel(
    const float* __restrict__ Yprev, const float* __restrict__ Wp, int kPrev,
    const float* __restrict__ U1, const float* __restrict__ W1,
    const float* __restrict__ U2, const float* __restrict__ W2,
    float* __restrict__ Yout)
{
  int wave = threadIdx.x >> 5, lane = threadIdx.x & 31;
  int r = lane & 15, half = lane >> 4, koff = 2 * half;
  int job = blockIdx.x * 8 + wave;
  int tm = job >> 3, n0 = (job & 7) * 64;
  const long rsY = (long)KB * DZ, rsU = (long)KB * DU;

  v8f acc[4]; zero4(acc);
  if (kPrev)
    wmma_bt<4>(acc, Yprev + (long)tm * 16 * rsY, rsY,
               Wp + (long)n0 * DZ, DZ, DZ, r, koff);
  wmma_bt<4>(acc, U1 + (long)tm * 16 * rsU, rsU,
             W1 + (long)n0 * DU, DU, DU, r, koff);
  wmma_bt<4>(acc, U2 + (long)tm * 16 * rsU, rsU,
             W2 + (long)n0 * DU, DU, DU, r, koff);
  store_strip<4>(Yout + (long)tm * 16 * rsY + n0, rsY, acc, r, half);
}

// ---- fill even steps: Y_j = Y_{j-1} @ A^T + U_j @ C^T (j = 2*blockIdx.z) --
__global__ void fill_even_kernel(float* __restrict__ Y,
                                 const float* __restrict__ A,
                                 const float* __restrict__ U,
                                 const float* __restrict__ C)
{
  int wave = threadIdx.x >> 5, lane = threadIdx.x & 31;
  int r = lane & 15, half = lane >> 4, koff = 2 * half;
  int job = blockIdx.x * 8 + wave;
  int tm = job >> 3, n0 = (job & 7) * 64;
  int j = 2 * blockIdx.z;
  const long rsY = (long)KB * DZ, rsU = (long)KB * DU;

  v8f acc[4]; zero4(acc);
  if (j > 0)
    wmma_bt<4>(acc, Y + (long)(j - 1) * DZ + (long)tm * 16 * rsY, rsY,
               A + (long)n0 * DZ, DZ, DZ, r, koff);
  wmma_bt<4>(acc, U + (long)j * DU + (long)tm * 16 * rsU, rsU,
             C + (long)n0 * DU, DU, DU, r, koff);
  store_strip<4>(Y + (long)j * DZ + (long)tm * 16 * rsY + n0, rsY,
                 acc, r, half);
}

// ---- triangular superblock aggregation: V[B,i] = sum_{q=0..i} M^q w[B,i-q] -
// w[B,i'] = Y row ((B*K2+i')*KB + KB-1). Rows with i-q < 0 masked by *0
// (no divergence: clamped address + multiply). jobs = 16*2*8 = 256.
__global__ void vtri_kernel(const float* __restrict__ Y,
                            const float* __restrict__ Mpow,
                            float* __restrict__ V)
{
  int wave = threadIdx.x >> 5, lane = threadIdx.x & 31;
  int r = lane & 15, half = lane >> 4, koff = 2 * half;
  int job = blockIdx.x * 8 + wave;
  int n0 = (job & 7) * 64;
  int it = (job >> 3) & 1;
  int B = job >> 4;

  v8f acc[4];
#pragma unroll
  for (int t = 0; t < 4; ++t)                     // q = 0 term (identity)
#pragma unroll
    for (int v = 0; v < 8; ++v) {
      int iv = it * 16 + v + 8 * half;
      acc[t][v] =
          Y[(((long)B * K2 + iv) * KB + (KB - 1)) * DZ + n0 + 16 * t + r];
    }
  int iA = it * 16 + r;                           // A-fragment row index i
  for (int q = 1; q < K2; ++q) {
    int i2 = iA - q;
    float mask = (i2 >= 0) ? 1.0f : 0.0f;
    int i2c = (i2 >= 0) ? i2 : 0;
    const float* ar = Y + (((long)B * K2 + i2c) * KB + (KB - 1)) * DZ + koff;
    const float* br =
        Mpow + (long)(q - 1) * DZ * DZ + (long)(n0 + r) * DZ + koff;
#pragma unroll 2
    for (int k = 0; k < DZ; k += 4) {
      v2f a = *(const v2f*)(ar + k);
      a = a * mask;
#pragma unroll
      for (int t = 0; t < 4; ++t) {
        v2f b = *(const v2f*)(br + (long)t * 16 * DZ + k);
        acc[t] = __builtin_amdgcn_wmma_f32_16x16x4_f32(false, a, false, b,
                                                       (short)0, acc[t],
                                                       false, false);
      }
    }
  }
#pragma unroll
  for (int t = 0; t < 4; ++t)
#pragma unroll
    for (int v = 0; v < 8; ++v) {
      int iv = it * 16 + v + 8 * half;
      V[((long)B * K2 + iv) * DZ + n0 + 16 * t + r] = acc[t][v];
    }
}

// ---- z0 = R @ (data[0] - mean) -> E row 0 ---------------------------------
__global__ void z0_kernel(const float* __restrict__ R,
                          const float* __restrict__ data,
                          const float* __restrict__ mean,
                          float* __restrict__ E)
{
  for (int m = threadIdx.x; m < DZ; m += 256) {
    const float* Rr = R + (long)m * DX;
    float acc = 0.f;
    for (int x = 0; x < DX; ++x) acc += Rr[x] * (data[x] - mean[x]);
    E[m] = acc;
  }
}

// ---- superblock carry scan: e_{B+1} = M^K2 e_B + v_{B,K2-1} ---------------
__global__ void escan_kernel(const float* __restrict__ Mpow,
                             const float* __restrict__ V,
                             float* __restrict__ E)
{
  __shared__ float e[DZ];
  int m = threadIdx.x;
  e[m] = E[m];
  __syncthreads();
  const float* M2 = Mpow + (long)(K2 - 1) * DZ * DZ;  // M^32
  for (int B = 0; B < NB2 - 1; ++B) {
    const float* Mr = M2 + (long)m * DZ;
    float acc = V[((long)B * K2 + (K2 - 1)) * DZ + m];
    for (int k = 0; k < DZ; ++k) acc += Mr[k] * e[k];  // e[k]: LDS broadcast
    __syncthreads();
    e[m] = acc;
    E[(long)(B + 1) * DZ + m] = acc;
    __syncthreads();
  }
}

// ---- block carries: c_{B*K2} = e_B ; c_{B*K2+i} = M^i e_B + v_{B,i-1} -----
__global__ void crecon_kernel(const float* __restrict__ Mpow,
                              const float* __restrict__ V,
                              const float* __restrict__ E,
                              float* __restrict__ Cbuf)
{
  __shared__ float e[DZ];
  int B = blockIdx.x, m = threadIdx.x;
  e[m] = E[(long)B * DZ + m];
  __syncthreads();
  Cbuf[((long)B * K2) * DZ + m] = e[m];
  for (int i = 1; i < K2; ++i) {
    const float* Mr = Mpow + (long)(i - 1) * DZ * DZ + (long)m * DZ;
    float acc = V[((long)B * K2 + i - 1) * DZ + m];
    for (int k = 0; k < DZ; ++k) acc += Mr[k] * e[k];
    Cbuf[((long)B * K2 + i) * DZ + m] = acc;
  }
}

// ---- fixup: Y[b*KB+j] += c_b @ (A^{j+1})^T  (j = blockIdx.z) --------------
__global__ void fixup_kernel(float* __restrict__ Y,
                             const float* __restrict__ Cbuf,
                             const float* __restrict__ Apow)
{
  int wave = threadIdx.x >> 5, lane = threadIdx.x & 31;
  int r = lane & 15, half = lane >> 4, koff = 2 * half;
  int job = blockIdx.x * 8 + wave;
  int tm = job >> 3, n0 = (job & 7) * 64;
  int j = blockIdx.z;
  const long rsY = (long)KB * DZ;
  float* Yt = Y + (long)j * DZ + (long)tm * 16 * rsY + n0;

  v8f acc[4];
#pragma unroll
  for (int t = 0; t < 4; ++t)
#pragma unroll
    for (int v = 0; v < 8; ++v)
      acc[t][v] = Yt[(long)(v + 8 * half) * rsY + 16 * t + r];
  wmma_bt<4>(acc, Cbuf + (long)tm * 16 * DZ, DZ,
             Apow + (long)j * DZ * DZ + (long)n0 * DZ, DZ, DZ, r, koff);
  store_strip<4>(Yt, rsY, acc, r, half);
}

// ---- output: out = Y @ B^T + mean -----------------------------------------
// Block = 8 waves sharing one 64-column n-strip of B; the strip is staged
// into LDS in 64x128 k-chunks with global_load_async_to_lds_b128 (ASYNCcnt),
// then each wave runs its WMMA strip with B fragments read from LDS.
// grid.x = 2048 blocks: low 3 bits = n-group, rest = m-tile block of 8.
__global__ void out_gemm_kernel(const float* __restrict__ Y,
                                const float* __restrict__ Bm,
                                const float* __restrict__ mean,
                                float* __restrict__ out)
{
  __shared__ float bs[64 * ROWP];
  int wave = threadIdx.x >> 5, lane = threadIdx.x & 31;
  int r = lane & 15, half = lane >> 4, koff = 2 * half;
  int n0 = (blockIdx.x & 7) * 64;
  int tm = (blockIdx.x >> 3) * 8 + wave;

  v8f acc[4];
#pragma unroll
  for (int t = 0; t < 4; ++t) {
    float mv = mean[n0 + 16 * t + r];
#pragma unroll
    for (int v = 0; v < 8; ++v) acc[t][v] = mv;
  }

  for (int kc = 0; kc < DZ; kc += 128) {
    // async-stage B rows [n0, n0+64), cols [kc, kc+128): 8 b128 per thread
#pragma unroll
    for (int rep = 0; rep < 8; ++rep) {
      int f = rep * 256 + threadIdx.x;     // float4 index: 2048 total
      int row = f >> 5, c4 = f & 31;       // 32 float4 per 128-float row
      const float* g = Bm + (long)(n0 + row) * DZ + kc + c4 * 4;
      unsigned la = (unsigned)(size_t)&bs[row * ROWP + c4 * 4];
      asm volatile("global_load_async_to_lds_b128 %0, %1, off"
                   :: "v"(la), "v"(g) : "memory");
    }
    asm volatile("s_wait_asynccnt 0x0" ::: "memory");
    __syncthreads();

    const float* arY = Y + ((long)tm * 16 + r) * DZ + kc + koff;
    const float* brL = &bs[r * ROWP + koff];
#pragma unroll 2
    for (int k = 0; k < 128; k += 4) {
      v2f a = *(const v2f*)(arY + k);
#pragma unroll
      for (int t = 0; t < 4; ++t) {
        v2f b = *(const v2f*)(brL + t * 16 * ROWP + k);
        acc[t] = __builtin_amdgcn_wmma_f32_16x16x4_f32(false, a, false, b,
                                                       (short)0, acc[t],
                                                       false, false);
      }
    }
    __syncthreads();
  }
  store_strip<4>(out + (long)tm * 16 * DX + n0, DX, acc, r, half);
}

// ---------------------------------------------------------------------------
extern "C" void kernel_launch(void* const* d_in, const int* in_sizes, int n_in,
                              void* d_out, int out_size, void* d_ws, size_t ws_size,
                              hipStream_t stream) {
  (void)in_sizes; (void)n_in; (void)out_size; (void)ws_size;
  const float* data   = (const float*)d_in[0];  // (T, DX)
  const float* inputs = (const float*)d_in[1];  // (T, DU)
  const float* mean   = (const float*)d_in[2];  // (1, DX)
  const float* A      = (const float*)d_in[3];  // (DZ, DZ)
  const float* Bm     = (const float*)d_in[4];  // (DX, DZ)
  const float* C      = (const float*)d_in[5];  // (DZ, DU)
  const float* R      = (const float*)d_in[6];  // (DZ, DX)
  float* out = (float*)d_out;

  float* ws   = (float*)d_ws;
  float* Y    = ws;                               // T*DZ          (64 MB)
  float* Apow = Y + (long)TSTEPS * DZ;            // 64*DZ*DZ: A^1..A^64
  float* Mpow = Apow + 64L * DZ * DZ;             // 32*DZ*DZ: M^1..M^32
  float* V    = Mpow + 32L * DZ * DZ;             // NB*DZ
  float* E    = V + (long)NB * DZ;                // NB2*DZ
  float* Cbuf = E + (long)NB2 * DZ;               // NB*DZ
  float* ACb  = Cbuf + (long)NB * DZ;             // DZ*DU  (A@C)

  const long MM = (long)DZ * DZ;
  dim3 blk(256);

  // A-power table: slot i = A^{i+1}, log-depth batched doubling
  hipMemcpyAsync(Apow, A, sizeof(float) * MM, hipMemcpyDeviceToDevice, stream);
  for (int rr = 1; rr <= 32; rr <<= 1)
    mm_batch_kernel<<<dim3(32, 1, rr), blk, 0, stream>>>(
        Apow, MM, Apow + (long)(rr - 1) * MM, Apow + (long)rr * MM, MM, DZ, DZ);

  // AC = A @ C (for stride-2 chaining); jobs = 32*4 = 128 -> 16 blocks
  mm_batch_kernel<<<dim3(16, 1, 1), blk, 0, stream>>>(A, 0, C, ACb, 0, DZ, DU);

  // M-power table: M = A^64, slot q-1 = M^q
  hipMemcpyAsync(Mpow, Apow + 63L * MM, sizeof(float) * MM,
                 hipMemcpyDeviceToDevice, stream);
  for (int rr = 1; rr <= 16; rr <<= 1)
    mm_batch_kernel<<<dim3(32, 1, rr), blk, 0, stream>>>(
        Mpow, MM, Mpow + (long)(rr - 1) * MM, Mpow + (long)rr * MM, MM, DZ, DZ);

  // e_0 = z0
  z0_kernel<<<1, blk, 0, stream>>>(R, data, mean, E);

  // within-block zero-init scans, odd steps via stride-2 chain:
  //   Y_1 = U_0@AC^T + U_1@C^T ; Y_j = Y_{j-2}@A2^T + U_{j-1}@AC^T + U_j@C^T
  chain_step_kernel<<<dim3(32), blk, 0, stream>>>(
      nullptr, nullptr, 0, inputs, ACb, inputs + DU, C, Y + DZ);
  for (int j = 3; j < KB; j += 2)
    chain_step_kernel<<<dim3(32), blk, 0, stream>>>(
        Y + (long)(j - 2) * DZ, Apow + 1L * MM, DZ,
        inputs + (long)(j - 1) * DU, ACb,
        inputs + (long)j * DU, C,
        Y + (long)j * DZ);

  // all even steps in one parallel launch
  fill_even_kernel<<<dim3(32, 1, KB / 2), blk, 0, stream>>>(Y, A, inputs, C);

  // superblock triangular aggregation, carry scan, block carries
  vtri_kernel<<<dim3(32), blk, 0, stream>>>(Y, Mpow, V);
  escan_kernel<<<1, dim3(512), 0, stream>>>(Mpow, V, E);
  crecon_kernel<<<dim3(NB2), dim3(512), 0, stream>>>(Mpow, V, E, Cbuf);

  // Y[t] += A^{j+1} c_b , then out = Y @ B^T + mean
  fixup_kernel<<<dim3(32, 1, KB), blk, 0, stream>>>(Y, Cbuf, Apow);
  out_gemm_kernel<<<dim3(2048), blk, 0, stream>>>(Y, Bm, mean, out);
}